// _RoIRingPooling_1726576857322
// MI455X (gfx1250) — compile-verified
//
#include <hip/hip_runtime.h>
#include <hip/hip_bf16.h>
#include <math.h>

// Problem constants (from reference)
#define PH_ 7
#define PW_ 7
#define C_  256
#define H_  50
#define W_  50
#define R_  128
#define BH_ 10   // ceil((H+1)/PH)+2
#define BW_ 10   // ceil((W+1)/PW)+2

// Tiling
#define CT      8     // channels per block
#define RMAX    40    // max staged region side (input dist gives <= ~30)
#define THREADS 256   // 8 wave32 waves

#if defined(__gfx1250__)
#  if __has_builtin(__builtin_amdgcn_tensor_load_to_lds)
#    define HAS_TDM 1
#  endif
#  if __has_builtin(__builtin_amdgcn_global_load_async_to_lds_b32)
#    define HAS_ASYNC_LDS 1
#  endif
#endif
#ifndef HAS_TDM
#  define HAS_TDM 0
#endif
#ifndef HAS_ASYNC_LDS
#  define HAS_ASYNC_LDS 0
#endif

typedef unsigned int v4u __attribute__((ext_vector_type(4)));
typedef int          v4i __attribute__((ext_vector_type(4)));
typedef int          v8i __attribute__((ext_vector_type(8)));

__device__ __forceinline__ int rnd_i(float v) {
    // reference: floor(v * SS + 0.5) with SS = 0.0625
    return (int)floorf(v * 0.0625f + 0.5f);
}
__device__ __forceinline__ int clampi(int v, int lo, int hi) {
    return v < lo ? lo : (v > hi ? hi : v);
}

__global__ __launch_bounds__(THREADS)
void roi_ring_pool_kernel(const float* __restrict__ feat,
                          const float* __restrict__ rois,
                          float* __restrict__ out)
{
    __shared__ float smem[CT * RMAX * RMAX];   // 51.2 KB

    const int r   = blockIdx.y;        // ROI
    const int c0  = blockIdx.x * CT;   // first channel of tile
    const int tid = threadIdx.x;

    // ---- ROI geometry (wave-uniform; compiler keeps it in scalar-float SALU) ----
    const float x1 = rois[r * 5 + 1], y1 = rois[r * 5 + 2];
    const float x2 = rois[r * 5 + 3], y2 = rois[r * 5 + 4];
    const float cx = (x1 + x2) * 0.5f, cy = (y1 + y2) * 0.5f;
    const float whalf = (x2 - x1) * 0.5f, hhalf = (y2 - y1) * 0.5f;
    const float SO = 1.8f, SI = 1.0f;

    const int hs  = rnd_i(cy - hhalf * SO), he  = rnd_i(cy + hhalf * SO);
    const int ws  = rnd_i(cx - whalf * SO), we  = rnd_i(cx + whalf * SO);
    const int hsi = rnd_i(cy - hhalf * SI), hei = rnd_i(cy + hhalf * SI);
    const int wsi = rnd_i(cx - whalf * SI), wei = rnd_i(cx + whalf * SI);

    const float roi_h = fmaxf((float)(he - hs + 1), 1.0f);
    const float roi_w = fmaxf((float)(we - ws + 1), 1.0f);
    const float bsh = roi_h / 7.0f, bsw = roi_w / 7.0f;

    // Bounding region over all bins: [h0,h1) x [w0,w1)
    const int h0 = clampi(hs, 0, H_);
    const int h1 = clampi((int)ceilf(7.0f * bsh) + hs, 0, H_);
    const int w0 = clampi(ws, 0, W_);
    const int w1 = clampi((int)ceilf(7.0f * bsw) + ws, 0, W_);
    const int regH = (h1 > h0) ? (h1 - h0) : 0;
    const int regW = (w1 > w0) ? (w1 - w0) : 0;
    const bool useLds = (regH <= RMAX) && (regW <= RMAX);
    const int chSize = regH * regW;

    // ---- Phase 1: stage region (CT channels) into LDS ----
    if (useLds) {
#if HAS_TDM
        // One TDM descriptor DMAs the whole CT x regH x regW tile into LDS.
        if (tid < 32) {                     // wave 0 only (TDM ignores EXEC)
            const unsigned int lds_base =
                (unsigned int)(unsigned long long)
                    ((__attribute__((address_space(3))) void*)smem);
            const unsigned long long gaddr =
                (unsigned long long)(const void*)(feat + (size_t)c0 * (H_ * W_)
                                                       + (size_t)h0 * W_ + w0);
            // D# group 0: count=1 | lds_addr | global_addr[56:0] | type=2
            v4u g0;
            g0.x = 1u;
            g0.y = lds_base;
            g0.z = (unsigned int)(gaddr & 0xFFFFFFFFull);
            g0.w = (unsigned int)((gaddr >> 32) & 0x1FFFFFFull) | (2u << 30);
            // D# group 1: mask=0, data_size=4B, dims/strides/tile dims
            v8i g1;
            g1[0] = (2 << 16);                                   // data_size=2 (4 bytes)
            g1[1] = (int)((unsigned)(regW & 0xFFFF) << 16);      // tensor_dim0[15:0]
            g1[2] = (int)(((unsigned)regW >> 16) |
                          ((unsigned)(regH & 0xFFFF) << 16));    // dim0 hi | tensor_dim1 lo
            g1[3] = (int)(((unsigned)regH >> 16) |
                          ((unsigned)(regW & 0xFFFF) << 16));    // dim1 hi | tile_dim0
            g1[4] = (int)((unsigned)(regH & 0xFFFF) |
                          ((unsigned)CT << 16));                 // tile_dim1 | tile_dim2
            g1[5] = W_;                                          // tensor_dim0_stride = 50
            g1[6] = (int)(((unsigned)(H_ * W_) & 0xFFFF) << 16); // dim1_stride[15:0] = 2500
            g1[7] = (int)((unsigned)(H_ * W_) >> 16);            // dim1_stride[47:16] = 0
            // D# group 2: tensor_dim2=CT; dim3/tile_dim3 unused
            v4i g2; g2[0] = CT; g2[1] = 0; g2[2] = 0; g2[3] = 0;
            v4i g3; g3[0] = 0;  g3[1] = 0; g3[2] = 0; g3[3] = 0;
            // Extra (6-arg toolchain form): unused trailing group, zero-filled
            v8i gx;
            gx[0] = 0; gx[1] = 0; gx[2] = 0; gx[3] = 0;
            gx[4] = 0; gx[5] = 0; gx[6] = 0; gx[7] = 0;

            __builtin_amdgcn_tensor_load_to_lds(g0, g1, g2, g3, gx, /*cpol=*/0);
#  if __has_builtin(__builtin_amdgcn_s_wait_tensorcnt)
            __builtin_amdgcn_s_wait_tensorcnt(0);
#  else
            asm volatile("s_wait_tensorcnt 0" ::: "memory");
#  endif
        }
#elif HAS_ASYNC_LDS
        const int total = CT * chSize;
        for (int i = tid; i < total; i += THREADS) {
            const int cl  = i / chSize;
            const int rem = i - cl * chSize;
            const int hh  = rem / regW;
            const int ww  = rem - hh * regW;
            const float* src = feat + (size_t)(c0 + cl) * (H_ * W_)
                                    + (size_t)(h0 + hh) * W_ + (w0 + ww);
            __builtin_amdgcn_global_load_async_to_lds_b32(
                (int*)src, (int*)&smem[i], /*offset=*/0, /*cpol=*/0);
        }
#  if __has_builtin(__builtin_amdgcn_s_wait_asynccnt)
        __builtin_amdgcn_s_wait_asynccnt(0);
#  else
        asm volatile("s_wait_asynccnt 0" ::: "memory");
#  endif
#else
        const int total = CT * chSize;
        for (int i = tid; i < total; i += THREADS) {
            const int cl  = i / chSize;
            const int rem = i - cl * chSize;
            const int hh  = rem / regW;
            const int ww  = rem - hh * regW;
            smem[i] = feat[(size_t)(c0 + cl) * (H_ * W_)
                           + (size_t)(h0 + hh) * W_ + (w0 + ww)];
        }
#endif
    }
    __syncthreads();

    // ---- Phase 2: per (channel, bin) ring max-pool ----
    const int outBase = (r * C_ + c0) * (PH_ * PW_);
    for (int i = tid; i < CT * PH_ * PW_; i += THREADS) {
        const int cl  = i / (PH_ * PW_);
        const int bin = i - cl * (PH_ * PW_);
        const int ph  = bin / PW_;
        const int pw  = bin - ph * PW_;

        int hstart = clampi((int)floorf((float)ph * bsh) + hs, 0, H_);
        int hend   = clampi((int)ceilf((float)(ph + 1) * bsh) + hs, 0, H_);
        int wstart = clampi((int)floorf((float)pw * bsw) + ws, 0, W_);
        int wend   = clampi((int)ceilf((float)(pw + 1) * bsw) + ws, 0, W_);
        // reference only enumerates BH/BW offsets per bin
        hend = min(hend, hstart + BH_);
        wend = min(wend, wstart + BW_);

        float mx  = -INFINITY;
        bool  any = false;
        if (useLds) {
            const float* base = &smem[cl * chSize];
            for (int h = hstart; h < hend; ++h) {
                const bool innh = (h > hsi) && (h < hei);
                const float* rowp = base + (h - h0) * regW - w0;
                for (int w = wstart; w < wend; ++w) {
                    if (innh && (w > wsi) && (w < wei)) continue;  // inner ring excluded
                    mx  = fmaxf(mx, rowp[w]);
                    any = true;
                }
            }
        } else {
            const float* base = feat + (size_t)(c0 + cl) * (H_ * W_);
            for (int h = hstart; h < hend; ++h) {
                const bool innh = (h > hsi) && (h < hei);
                for (int w = wstart; w < wend; ++w) {
                    if (innh && (w > wsi) && (w < wei)) continue;
                    mx  = fmaxf(mx, base[h * W_ + w]);
                    any = true;
                }
            }
        }
        out[outBase + cl * (PH_ * PW_) + bin] = any ? mx : 0.0f;
    }
}

extern "C" void kernel_launch(void* const* d_in, const int* in_sizes, int n_in,
                              void* d_out, int out_size, void* d_ws, size_t ws_size,
                              hipStream_t stream) {
    const float* feat = (const float*)d_in[0];   // (1, 256, 50, 50) fp32
    const float* rois = (const float*)d_in[1];   // (128, 5) fp32
    float* out = (float*)d_out;                  // (128, 256, 7, 7) fp32

    dim3 grid(C_ / CT, R_);
    roi_ring_pool_kernel<<<grid, THREADS, 0, stream>>>(feat, rois, out);
}